// FourierLatentLayer_76596446757533
// MI455X (gfx1250) — compile-verified
//
#include <hip/hip_runtime.h>
#include <hip/hip_bf16.h>

typedef float v2f __attribute__((ext_vector_type(2)));
typedef float v8f __attribute__((ext_vector_type(8)));

#define NFREQ 16
#define INV_TWO_PI 0.15915494309189535f   // 1/(2*pi)

// ---------------------------------------------------------------------------
// Setup kernel (one wave): fold the sin/cos coefficient pair into an
// amplitude+phase form, fold the frange normalization into the amplitudes and
// offset, and pre-scale frequency/phase by 1/(2*pi) so the main kernel can
// feed the raw hardware v_sin_f32 (which computes sin(2*pi*x)) directly.
//   ws[0..15]  = Rs[f]     = sqrt(s^2+c^2)/frange
//   ws[16..31] = ph_rev[f] = atan2(c, s)/(2*pi)
//   ws[32]     = c0        = offset/frange
//   ws[33..48] = fr_rev[f] = freqs[f]/(2*pi)
// ---------------------------------------------------------------------------
__global__ void fourier_setup_kernel(const float* __restrict__ freqs,
                                     const float* __restrict__ offset,
                                     const float* __restrict__ sc,
                                     const float* __restrict__ cc,
                                     float* __restrict__ ws) {
    int l = threadIdx.x;
    float ssum = 0.0f;
#pragma unroll
    for (int f = 0; f < NFREQ; ++f)
        ssum += fabsf(sc[f]) + fabsf(cc[f]);
    float inv = 1.0f / fmaxf(1.0f, ssum + 1e-8f);
    if (l < NFREQ) {
        float s = sc[l], c = cc[l];
        ws[l]             = sqrtf(s * s + c * c) * inv;     // Rs
        ws[NFREQ + l]     = atan2f(c, s) * INV_TWO_PI;      // phase (revs)
        ws[2 * NFREQ + 1 + l] = freqs[l] * INV_TWO_PI;      // freq (revs)
    }
    if (l == 0) ws[2 * NFREQ] = offset[0] * inv;            // c0
}

// ---------------------------------------------------------------------------
// Main kernel: out[i] = c0 + sum_f Rs[f] * sin(t[i]*freqs[f] + phi[f])
// Reduction over f via chained V_WMMA_F32_16X16X4_F32:
//   B (4x16)  = sin values, computed per-lane directly in B layout
//               (lanes 0-15: K=0,1 ; lanes 16-31: K=2,3)
//   A (16x4)  = Rs replicated in every row -> every row of D is the result
//   C init    = c0 broadcast (offset folded into accumulator)
// Two accumulators per wave cover 32 timesteps; the result for timestep
// (lane&15) lands in D vgpr0 of every lane -> branchless select + one fully
// coalesced 32-lane store. Inner loop cost per 32 outputs:
//   16 v_fma (VALU) + 16 v_sin (TRANS) + 8 v_wmma (matrix pipe).
// ---------------------------------------------------------------------------
__global__ void fourier_latent_wmma_kernel(const float* __restrict__ t,
                                           const float* __restrict__ ws,
                                           float* __restrict__ out,
                                           int n) {
    const int lane   = threadIdx.x & 31;
    const int waveId = threadIdx.x >> 5;
    const int wavesPerBlock = blockDim.x >> 5;
    const int half = (lane >> 4) & 1;        // 0: lanes 0-15, 1: lanes 16-31

    // Per-lane constants for the 4 chained K=4 WMMA chunks.
    float a0[4], a1[4], fr0[4], fr1[4], ph0[4], ph1[4];
#pragma unroll
    for (int c = 0; c < 4; ++c) {
        int k0 = 4 * c + 2 * half;           // K handled by this lane-half
        int k1 = k0 + 1;
        a0[c]  = ws[k0];                  a1[c]  = ws[k1];                  // Rs
        ph0[c] = ws[NFREQ + k0];          ph1[c] = ws[NFREQ + k1];          // phase (revs)
        fr0[c] = ws[2 * NFREQ + 1 + k0];  fr1[c] = ws[2 * NFREQ + 1 + k1];  // freq (revs)
    }
    const float c0 = ws[2 * NFREQ];

    long long base   = ((long long)blockIdx.x * wavesPerBlock + waveId) * 32;
    const long long stride = (long long)gridDim.x * wavesPerBlock * 32;

    for (; base + 32 <= n; base += stride) {
        // 16 distinct timesteps per group, broadcast across lane halves.
        float tA = t[base +      (lane & 15)];
        float tB = t[base + 16 + (lane & 15)];

        v8f accA, accB;
#pragma unroll
        for (int i = 0; i < 8; ++i) { accA[i] = c0; accB[i] = c0; }

#pragma unroll
        for (int c = 0; c < 4; ++c) {
            v2f a  = { a0[c], a1[c] };
            // raw v_sin_f32: computes sin(2*pi*x); freq/phase pre-scaled.
            v2f bA = { __builtin_amdgcn_sinf(__builtin_fmaf(tA, fr0[c], ph0[c])),
                       __builtin_amdgcn_sinf(__builtin_fmaf(tA, fr1[c], ph1[c])) };
            v2f bB = { __builtin_amdgcn_sinf(__builtin_fmaf(tB, fr0[c], ph0[c])),
                       __builtin_amdgcn_sinf(__builtin_fmaf(tB, fr1[c], ph1[c])) };
            accA = __builtin_amdgcn_wmma_f32_16x16x4_f32(
                false, a, false, bA, (short)0, accA, false, false);
            accB = __builtin_amdgcn_wmma_f32_16x16x4_f32(
                false, a, false, bB, (short)0, accB, false, false);
        }

        // D vgpr0: every lane holds result for timestep (lane&15) of its group.
        float val = half ? accB[0] : accA[0];
        out[base + lane] = val;
    }
}

extern "C" void kernel_launch(void* const* d_in, const int* in_sizes, int n_in,
                              void* d_out, int out_size, void* d_ws, size_t ws_size,
                              hipStream_t stream) {
    const float* t      = (const float*)d_in[0];
    const float* freqs  = (const float*)d_in[1];
    const float* offset = (const float*)d_in[2];
    const float* sc     = (const float*)d_in[3];
    const float* cc     = (const float*)d_in[4];
    float*       out    = (float*)d_out;
    float*       ws     = (float*)d_ws;
    const int n = in_sizes[0];

    fourier_setup_kernel<<<1, 32, 0, stream>>>(freqs, offset, sc, cc, ws);

    const int block = 256;   // 8 waves
    const int grid  = 2048;  // 512K outputs per sweep -> 8 grid-stride sweeps
    fourier_latent_wmma_kernel<<<grid, block, 0, stream>>>(t, ws, out, n);
}